// ClassicalRBFKernel_65481071402788
// MI455X (gfx1250) — compile-verified
//
#include <hip/hip_runtime.h>

typedef float v2f __attribute__((ext_vector_type(2)));
typedef float v8f __attribute__((ext_vector_type(8)));

#define DIM_N 8192
#define DIM_D 64

// Each wave computes a 32x32 output tile via 2x2 WMMA 16x16 tiles,
// K=64 swept in 16 steps of V_WMMA_F32_16X16X4_F32.
// Block = 256 threads = 8 waves (2 in M x 4 in N) -> 64x128 per block.
__global__ __launch_bounds__(256) void rbf_wmma_f32_kernel(
    const float* __restrict__ X, const float* __restrict__ Y,
    float* __restrict__ out) {
  const int lane = threadIdx.x & 31;
  const int wave = threadIdx.x >> 5;
  const int half = lane >> 4;   // which half-wave (K split)
  const int r    = lane & 15;   // row/col within a 16-tile

  const int waveM = wave >> 2;  // 0..1
  const int waveN = wave & 3;   // 0..3
  const int rowBase = blockIdx.y * 64  + waveM * 32;
  const int colBase = blockIdx.x * 128 + waveN * 32;

  // ---- Load A fragments (two 16-row tiles of X) and B fragments (two
  // 16-col tiles of Y). WMMA f32 16x16x4 layout: lane (half,r) holds
  // K = 4*s + 2*half + {0,1} of row r for step s -> one float2 per step.
  v2f a[2][16], b[2][16];
  float sa[2] = {0.f, 0.f};     // per-lane partial ||x_row||^2 (half the K's)
  float sb[2] = {0.f, 0.f};     // per-lane partial ||y_col||^2
#pragma unroll
  for (int t = 0; t < 2; ++t) {
    const float* xr = X + (size_t)(rowBase + t * 16 + r) * DIM_D + 2 * half;
    const float* yr = Y + (size_t)(colBase + t * 16 + r) * DIM_D + 2 * half;
#pragma unroll
    for (int s = 0; s < 16; ++s) {
      v2f av = *(const v2f*)(xr + 4 * s);
      v2f bv = *(const v2f*)(yr + 4 * s);
      a[t][s] = av;
      b[t][s] = bv;
      sa[t] += av.x * av.x + av.y * av.y;
      sb[t] += bv.x * bv.x + bv.y * bv.y;
    }
  }
  // Merge the two half-wave K partitions: full row/col sum of squares.
#pragma unroll
  for (int t = 0; t < 2; ++t) {
    sa[t] += __shfl_xor(sa[t], 16, 32);
    sb[t] += __shfl_xor(sb[t], 16, 32);
  }

  // ---- GEMM accumulation on the matrix pipe.
  v8f acc[2][2] = {};
#pragma unroll
  for (int s = 0; s < 16; ++s) {
#pragma unroll
    for (int i = 0; i < 2; ++i) {
#pragma unroll
      for (int j = 0; j < 2; ++j) {
        acc[i][j] = __builtin_amdgcn_wmma_f32_16x16x4_f32(
            false, a[i][s], false, b[j][s], (short)0, acc[i][j], false, false);
      }
    }
  }

  // ---- Epilogue: exp(-max(||x||^2 + ||y||^2 - 2 x.y, 0)), coalesced store.
  // C/D layout: VGPR v, lanes 0-15 -> (M=v, N=lane); lanes 16-31 -> (M=v+8).
#pragma unroll
  for (int i = 0; i < 2; ++i) {
#pragma unroll
    for (int j = 0; j < 2; ++j) {
      const float y2 = sb[j];  // this lane's column is N = r
#pragma unroll
      for (int v = 0; v < 8; ++v) {
        const int mloc = v + 8 * half;
        const float x2 = __shfl(sa[i], mloc, 32);  // row sumsq lives in lane mloc
        float sq = x2 + y2 - 2.0f * acc[i][j][v];
        sq = fmaxf(sq, 0.0f);
        const int row = rowBase + i * 16 + mloc;
        const int col = colBase + j * 16 + r;
        out[(size_t)row * DIM_N + col] = __expf(-sq);
      }
    }
  }
}

extern "C" void kernel_launch(void* const* d_in, const int* in_sizes, int n_in,
                              void* d_out, int out_size, void* d_ws, size_t ws_size,
                              hipStream_t stream) {
  (void)in_sizes; (void)n_in; (void)d_ws; (void)ws_size; (void)out_size;
  const float* X = (const float*)d_in[0];
  const float* Y = (const float*)d_in[1];
  float* out = (float*)d_out;

  dim3 block(256, 1, 1);
  dim3 grid(DIM_N / 128, DIM_N / 64, 1);  // x: col blocks (128 wide), y: row blocks (64 tall)
  rbf_wmma_f32_kernel<<<grid, block, 0, stream>>>(X, Y, out);
}